// SigLayer_40029095198921
// MI455X (gfx1250) — compile-verified
//
#include <hip/hip_runtime.h>

typedef __attribute__((ext_vector_type(2))) float v2f;
typedef __attribute__((ext_vector_type(8))) float v8f;

#define NB 32
#define LPATH 2048
#define DDIM 8
#define TSTEPS 2047          // LPATH - 1 increments
#define SIG 584              // 8 + 64 + 512
#define L3OFF 72             // offset of level-3 block in a signature row
#define TILE 16
#define NTILES 128           // ceil(2047 / 16)
#define WPAD 520             // padded row stride for W in LDS (bank-conflict free)

// ---------------------------------------------------------------------------
// K1: per-batch sequential scan of levels 1 & 2 (cheap: 72 values/step) plus
// the running level-3 prefix; spills level-3 carries at each 16-step tile
// boundary into the (not yet written) level-3 slots of d_out.
// ---------------------------------------------------------------------------
__global__ __launch_bounds__(512) void sig_scan_low(const float* __restrict__ path,
                                                    float* __restrict__ out) {
  const int b   = blockIdx.x;
  const int tid = threadIdx.x;
  __shared__ float zsh[DDIM], a1sh[DDIM], a2sh[64];
  if (tid < DDIM) { zsh[tid] = 0.f; a1sh[tid] = 0.f; }
  if (tid < 64)   a2sh[tid] = 0.f;

  float acc3 = 0.f;                         // this thread's element of A3 prefix
  const int ij = tid >> 3, kk = tid & 7;    // level-3 element (i*8+j, k)
  const int i3 = ij >> 3, j3 = ij & 7;
  const size_t pbase = (size_t)b * LPATH * DDIM;
  const size_t obase = (size_t)b * TSTEPS * SIG;
  __syncthreads();

  for (int t = 0; t < TSTEPS; ++t) {
    if ((t & (TILE - 1)) == 0) {
      // carry = A3(t-1) for the tile starting at t (coalesced 512-float store)
      out[obase + (size_t)t * SIG + L3OFF + tid] = acc3;
    }
    if (tid < DDIM) {
      zsh[tid] = path[pbase + (size_t)(t + 1) * DDIM + tid] -
                 path[pbase + (size_t)t * DDIM + tid];
    }
    __syncthreads();

    // level-3 increment uses OLD A1/A2:  M3 = (A2 + (A1 + z/3) (x) z / 2) (x) z
    {
      float zi = zsh[i3], zj = zsh[j3], zk = zsh[kk];
      float b2 = a2sh[ij] + (a1sh[i3] + zi * (1.f / 3.f)) * zj * 0.5f;
      acc3 += b2 * zk;
    }
    float newa1 = 0.f, newa2 = 0.f;
    if (tid < DDIM) newa1 = a1sh[tid] + zsh[tid];
    if (tid < 64)
      newa2 = a2sh[tid] + (a1sh[tid >> 3] + zsh[tid >> 3] * 0.5f) * zsh[tid & 7];
    __syncthreads();

    if (tid < DDIM) { a1sh[tid] = newa1; out[obase + (size_t)t * SIG + tid] = newa1; }
    if (tid < 64)   { a2sh[tid] = newa2; out[obase + (size_t)t * SIG + 8 + tid] = newa2; }
    __syncthreads();
  }
}

// ---------------------------------------------------------------------------
// K2: one block per (batch, 16-step tile). Builds W[16][512] (level-3 per-step
// increments) in LDS, computes the in-tile inclusive prefix P = L * W with
// V_WMMA_F32_16X16X4_F32 (L = 16x16 lower-triangular ones). The per-column
// tile carry is folded into the WMMA accumulator's initial value.
// ---------------------------------------------------------------------------
__global__ __launch_bounds__(256) void sig_tile_l3(const float* __restrict__ path,
                                                   float* __restrict__ out) {
  const int tile = blockIdx.x;
  const int b    = blockIdx.y;
  const int tid  = threadIdx.x;
  const int t0   = tile * TILE;

  __shared__ float zsh[TILE][DDIM];
  __shared__ float a1p[TILE][DDIM];
  __shared__ float a2p[TILE][64];
  __shared__ float W[TILE][WPAD];
  __shared__ float carry[512];

  const size_t pbase = (size_t)b * LPATH * DDIM;
  const size_t obase = (size_t)b * TSTEPS * SIG;

  if (tid < 128) {
    int s = tid >> 3, k = tid & 7;
    int t = t0 + s;
    bool valid = t < TSTEPS;
    zsh[s][k] = valid ? (path[pbase + (size_t)(t + 1) * DDIM + k] -
                         path[pbase + (size_t)t * DDIM + k]) : 0.f;
    a1p[s][k] = (valid && t > 0) ? out[obase + (size_t)(t - 1) * SIG + k] : 0.f;
  }
  for (int r = 0; r < 4; ++r) {
    int idx = tid + 256 * r;            // 0..1023
    int s = idx >> 6, j = idx & 63;
    int t = t0 + s;
    a2p[s][j] = (t < TSTEPS && t > 0) ? out[obase + (size_t)(t - 1) * SIG + 8 + j] : 0.f;
  }
  // carry written by K1 into the level-3 slot of row t0 (read before overwrite)
  carry[tid]       = out[obase + (size_t)t0 * SIG + L3OFF + tid];
  carry[tid + 256] = out[obase + (size_t)t0 * SIG + L3OFF + tid + 256];
  __syncthreads();

  // W[s][(i*8+j)*8+k] = (A2p[ij] + (A1p[i] + z_i/3) * z_j / 2) * z_k
  for (int r = 0; r < 32; ++r) {
    int idx = r * 256 + tid;            // 0..8191
    int s = idx >> 9, rem = idx & 511;
    int ij = rem >> 3, k = rem & 7, i = ij >> 3, j = ij & 7;
    W[s][rem] = (a2p[s][ij] + (a1p[s][i] + zsh[s][i] * (1.f / 3.f)) * zsh[s][j] * 0.5f)
                * zsh[s][k];
  }
  __syncthreads();

  // Triangular prefix via f32 WMMA: 8 waves x 4 column-chunks of 16.
  const int wave = tid >> 5;
  const int lane = tid & 31;
  const int m    = lane & 15;           // A row / B,D column within chunk
  const bool hi  = lane >= 16;

  // A operand (L chunks) depends only on the lane -> build once, reuse 4x.
  v2f amat[4];
#pragma unroll
  for (int q = 0; q < 4; ++q) {
    int k0 = q * 4 + (hi ? 2 : 0);
    amat[q].x = (k0 <= m) ? 1.f : 0.f;      // L[m][k0]
    amat[q].y = (k0 + 1 <= m) ? 1.f : 0.f;  // L[m][k0+1]
  }

  const bool full = (t0 + TILE) <= TSTEPS;  // uniform: only last tile is partial

  for (int cc = 0; cc < 4; ++cc) {
    const int n0  = (wave * 4 + cc) * 16;
    const int col = n0 + m;

    // Fold the per-column carry into the accumulator: every C/D component in
    // this lane maps to the same column, so broadcast carry[col].
    const float cv = carry[col];
    v8f acc = {cv, cv, cv, cv, cv, cv, cv, cv};

#pragma unroll
    for (int q = 0; q < 4; ++q) {
      int k0 = q * 4 + (hi ? 2 : 0);
      v2f bm;
      bm.x = W[k0][col];
      bm.y = W[k0 + 1][col];
      acc = __builtin_amdgcn_wmma_f32_16x16x4_f32(
          /*neg_a=*/false, amat[q], /*neg_b=*/false, bm,
          /*c_mod=*/(short)0, acc, /*reuse_a=*/false, /*reuse_b=*/false);
    }

    // Row (v + 8*hi) of this tile, column col. Base pointer per lane; the 8
    // row stores become constant-offset global_store_b32s (stride SIG*4 B).
    float* p = out + obase + (size_t)(t0 + (hi ? 8 : 0)) * SIG + L3OFF + col;
    if (full) {
#pragma unroll
      for (int v = 0; v < 8; ++v) p[(size_t)v * SIG] = acc[v];
    } else {
      const int lim = TSTEPS - t0 - (hi ? 8 : 0);  // rows valid in this half
#pragma unroll
      for (int v = 0; v < 8; ++v)
        if (v < lim) p[(size_t)v * SIG] = acc[v];
    }
  }
}

extern "C" void kernel_launch(void* const* d_in, const int* in_sizes, int n_in,
                              void* d_out, int out_size, void* d_ws, size_t ws_size,
                              hipStream_t stream) {
  (void)in_sizes; (void)n_in; (void)out_size; (void)d_ws; (void)ws_size;
  const float* path = (const float*)d_in[0];
  float* out = (float*)d_out;

  // Pass 1: levels 1-2 prefix + level-3 tile carries (stream-ordered before K2).
  sig_scan_low<<<NB, 512, 0, stream>>>(path, out);
  // Pass 2: 4096 independent tile blocks; WMMA triangular prefix for level 3.
  sig_tile_l3<<<dim3(NTILES, NB), 256, 0, stream>>>(path, out);
}